// SimTALayer_27315992002802
// MI455X (gfx1250) — compile-verified
//
#include <hip/hip_runtime.h>
#include <math.h>

typedef __attribute__((ext_vector_type(2))) float v2f;
typedef __attribute__((ext_vector_type(8))) float v8f;

#define B_    4
#define T_    4096
#define DIN   64
#define DM    256
#define BT    (B_ * T_)
#define CHUNK 128
#define NC    (T_ / CHUNK)

// ---------------------------------------------------------------------------
// Kernel 1: h = elu(x @ W^T + bias)   via V_WMMA_F32_16X16X4_F32
//   x: [BT, 64] row-major, W: [256, 64] row-major (so B-matrix = W^T)
//   One 16x16 output tile per wave; K=64 as 16 chained k=4 WMMAs.
// ---------------------------------------------------------------------------
__global__ __launch_bounds__(256) void gemm_elu_wmma(
    const float* __restrict__ x, const float* __restrict__ W,
    const float* __restrict__ bias, float* __restrict__ h)
{
    const int wave = threadIdx.x >> 5;
    const int lane = threadIdx.x & 31;
    const int tile = blockIdx.x * 8 + wave;   // 8 waves/block, 1 tile/wave
    const int mt   = tile >> 4;               // 1024 M-tiles (BT/16)
    const int nt   = tile & 15;               // 16  N-tiles (DM/16)
    const int half = lane >> 4;               // lane group 0/1
    const int l16  = lane & 15;
    const int kb   = half * 2;                // K sub-slot inside k=4 slab

    const float* __restrict__ xr = x + (size_t)(mt * 16 + l16) * DIN; // A row
    const float* __restrict__ wr = W + (size_t)(nt * 16 + l16) * DIN; // B col (=W row)

    v8f acc = {};
#pragma unroll
    for (int k = 0; k < DIN; k += 4) {
        // A 16x4 f32: VGPR0=K(kb), VGPR1=K(kb+1)  -> contiguous float2
        v2f a = *(const v2f*)(xr + k + kb);
        // B 4x16 f32: mirrored striping (half selects K pair)
        v2f b = *(const v2f*)(wr + k + kb);
        acc = __builtin_amdgcn_wmma_f32_16x16x4_f32(
            /*neg_a=*/false, a, /*neg_b=*/false, b,
            /*c_mod=*/(short)0, acc, /*reuse_a=*/false, /*reuse_b=*/false);
    }

    // Epilogue: C/D layout -> VGPR r holds (M = r + half*8, N = l16)
    const int ng = nt * 16 + l16;
    const float bn = bias[ng];
#pragma unroll
    for (int r = 0; r < 8; ++r) {
        const int mg = mt * 16 + r + half * 8;
        float v = acc[r] + bn;
        v = v > 0.0f ? v : expm1f(v);         // elu, alpha=1
        h[(size_t)mg * DM + ng] = v;
    }
}

// ---------------------------------------------------------------------------
// Chunked exponential-moving-average scan (exact softmax-attention collapse):
//   d[t] = exp(-relu(lamb)*tau[t]);  V = d*V + h;  S = d*S + 1;  out = V/S
// ---------------------------------------------------------------------------

// Pass 1: per (b, chunk) local scan with zero init; emit V_end, S_end, P=prod(d)
__global__ __launch_bounds__(256) void scan_pass1(
    const float* __restrict__ h, const float* __restrict__ tau,
    const float* __restrict__ lamb_p,
    float* __restrict__ S_loc, float* __restrict__ P_arr,
    float* __restrict__ V_loc)
{
    const int b = blockIdx.x / NC;
    const int c = blockIdx.x % NC;
    const int m = threadIdx.x;                // channel
    const float lr = fmaxf(lamb_p[0], 0.0f);
    const int t0 = c * CHUNK;

    const float* __restrict__ hp = h + ((size_t)b * T_ + t0) * DM + m;
    const float* __restrict__ tp = tau + (size_t)b * T_ + t0;

    float v = 0.0f, s = 0.0f, p = 1.0f;
    for (int t = 0; t < CHUNK; ++t) {
        const float d = __expf(-lr * tp[t]); // broadcast scalar, cached
        v = d * v + hp[(size_t)t * DM];      // coalesced across 256 channels
        s = d * s + 1.0f;
        p *= d;
    }
    V_loc[((size_t)b * NC + c) * DM + m] = v;
    if (m == 0) { S_loc[b * NC + c] = s; P_arr[b * NC + c] = p; }
}

// Pass 2: sequential cross-chunk prefix (tiny: NC=32 steps, 1 block per batch)
__global__ __launch_bounds__(256) void scan_pass2(
    const float* __restrict__ S_loc, const float* __restrict__ P_arr,
    const float* __restrict__ V_loc,
    float* __restrict__ S_init, float* __restrict__ V_init)
{
    const int b = blockIdx.x;
    const int m = threadIdx.x;
    float vinit = 0.0f, sinit = 0.0f;
    for (int c = 0; c < NC; ++c) {
        const size_t idx = ((size_t)b * NC + c) * DM + m;
        V_init[idx] = vinit;
        if (m == 0) S_init[b * NC + c] = sinit;
        const float p = P_arr[b * NC + c];
        vinit = p * vinit + V_loc[idx];
        if (m == 0) sinit = p * sinit + S_loc[b * NC + c];
    }
}

// Pass 3: re-scan each chunk with its true init; write out = V/S in place
// (each thread reads exactly the h element it overwrites -> alias-safe)
__global__ __launch_bounds__(256) void scan_pass3(
    float* __restrict__ hout, const float* __restrict__ tau,
    const float* __restrict__ lamb_p,
    const float* __restrict__ S_init, const float* __restrict__ V_init)
{
    const int b = blockIdx.x / NC;
    const int c = blockIdx.x % NC;
    const int m = threadIdx.x;
    const float lr = fmaxf(lamb_p[0], 0.0f);
    const int t0 = c * CHUNK;

    float* __restrict__ hp = hout + ((size_t)b * T_ + t0) * DM + m;
    const float* __restrict__ tp = tau + (size_t)b * T_ + t0;

    float v = V_init[((size_t)b * NC + c) * DM + m];
    float s = S_init[b * NC + c];
    for (int t = 0; t < CHUNK; ++t) {
        const float d = __expf(-lr * tp[t]);
        v = d * v + hp[(size_t)t * DM];
        s = d * s + 1.0f;
        hp[(size_t)t * DM] = v / s;
    }
}

// ---------------------------------------------------------------------------
extern "C" void kernel_launch(void* const* d_in, const int* in_sizes, int n_in,
                              void* d_out, int out_size, void* d_ws, size_t ws_size,
                              hipStream_t stream) {
    const float* x    = (const float*)d_in[0];  // [4,4096,64]
    const float* tau  = (const float*)d_in[1];  // [4,4096]
    const float* W    = (const float*)d_in[2];  // [256,64]
    const float* bias = (const float*)d_in[3];  // [256]
    const float* lamb = (const float*)d_in[4];  // scalar
    // d_in[5] = beta: provably cancels in the row-softmax, unused.
    float* out = (float*)d_out;                 // holds h, overwritten by out

    float* ws     = (float*)d_ws;               // ~264 KB total
    float* S_loc  = ws;                         // [B*NC]
    float* P_arr  = ws + (size_t)B_ * NC;       // [B*NC]
    float* S_init = ws + (size_t)2 * B_ * NC;   // [B*NC]
    float* V_loc  = ws + (size_t)3 * B_ * NC;   // [B*NC*DM]
    float* V_init = V_loc + (size_t)B_ * NC * DM;

    // 1) h = elu(x W^T + b): (BT/16)*(DM/16) tiles / 8 waves per block
    gemm_elu_wmma<<<(BT / 16) * (DM / 16) / 8, 256, 0, stream>>>(x, W, bias, out);
    // 2-4) chunked EMA scan == exact causal time-decay softmax attention
    scan_pass1<<<B_ * NC, 256, 0, stream>>>(out, tau, lamb, S_loc, P_arr, V_loc);
    scan_pass2<<<B_, 256, 0, stream>>>(S_loc, P_arr, V_loc, S_init, V_init);
    scan_pass3<<<B_ * NC, 256, 0, stream>>>(out, tau, lamb, S_init, V_init);
}